// CNNcond_9723805958518
// MI455X (gfx1250) — compile-verified
//
#include <hip/hip_runtime.h>
#include <stdint.h>

// out[b,i] = bias + sum_{k<16,d<512} x[b,i+k,d] * w[k,d]   (zero-pad 15 frames at end)
// GEMM  Y = X * W^T  (N = 16 taps) on V_WMMA_F32_16X16X4_F32, diagonal epilogue.
// X is staged global->LDS with GLOBAL_LOAD_ASYNC_TO_LDS_B128, double-buffered,
// pipelined one chunk ahead with s_wait_asynccnt partial waits.

typedef float v2f __attribute__((ext_vector_type(2)));
typedef float v4f __attribute__((ext_vector_type(4)));
typedef float v8f __attribute__((ext_vector_type(8)));
typedef int   v4i __attribute__((vector_size(16)));   // matches builtin's pointee type

constexpr int S    = 4096;      // sequence length
constexpr int D    = 512;       // channels (GEMM contraction dim)
constexpr int KT   = 16;        // taps (GEMM N dim)
constexpr int TM   = 64;        // outputs per workgroup
constexpr int ROWS = 80;        // X rows staged = TM + 16 (diagonal lookahead)
constexpr int DC   = 32;        // channel chunk per pipeline stage
constexpr int NCH  = D / DC;    // 16 chunks
constexpr int SXW  = 36;        // ldsX row stride: 36*r mod 64 = 4*(9r mod 16) -> conflict-free
constexpr int XBUF = ROWS * SXW;
constexpr int SWW  = 516;       // ldsW row stride; 516 % 64 == 4 -> conflict-free
constexpr int SYW  = 18;        // ldsY row stride; conflict-free diagonal gather
constexpr int APT  = 5;         // async b128 per thread per chunk = 80*32/4/128

#if __has_builtin(__builtin_amdgcn_global_load_async_to_lds_b128)
#define ASYNC_COPY_B128(dst_lds, src_glb)                                        \
  __builtin_amdgcn_global_load_async_to_lds_b128(                                \
      (__attribute__((address_space(1))) v4i*)(uintptr_t)(src_glb),              \
      (__attribute__((address_space(3))) v4i*)(uintptr_t)(dst_lds), 0, 0)
#else
#define ASYNC_COPY_B128(dst_lds, src_glb)                                        \
  asm volatile("global_load_async_to_lds_b128 %0, %1, off"                       \
               :: "v"((uint32_t)(uintptr_t)(dst_lds)),                           \
                  "v"((const void*)(src_glb)) : "memory")
#endif

#if __has_builtin(__builtin_amdgcn_s_wait_asynccnt)
#define WAIT_ASYNC(n) __builtin_amdgcn_s_wait_asynccnt(n)
#else
#define WAIT_ASYNC(n) asm volatile("s_wait_asynccnt %0" :: "n"(n) : "memory")
#endif

__global__ __launch_bounds__(128) void convk16_wmma_f32_async(
    const float* __restrict__ x,     // [B, S, D]
    const float* __restrict__ w,     // [KT*D] (k-major)
    const float* __restrict__ bias,  // [1]
    float* __restrict__ out)         // [B, S]
{
  __shared__ float ldsX[2 * XBUF];      // 23040 B (double buffer)
  __shared__ float ldsW[KT * SWW];      // 33024 B
  __shared__ float ldsY[4 * 32 * SYW];  //  9216 B
  __shared__ float ldsTrash[4];         //    16 B sink for zero-padded lanes

  const int tid  = threadIdx.x;
  const int wave = tid >> 5;
  const int lane = tid & 31;
  const int lm   = lane & 15;
  const int lh   = lane >> 4;

  const int b  = blockIdx.y;
  const int i0 = blockIdx.x * TM;
  const float* xb = x + (size_t)b * S * D;

  // Pre-zero X buffers once if this block touches the zero-padded tail; pad rows
  // are never overwritten because their async lanes are redirected to ldsTrash.
  if (i0 + ROWS > S) {
    for (int t = tid; t < 2 * XBUF; t += 128) ldsX[t] = 0.f;
  }

  // Stage filter [16][512] -> LDS once (coalesced b128).
  #pragma unroll
  for (int j = 0; j < 16; ++j) {
    int q = tid + 128 * j, row = q >> 7, c4 = q & 127;
    *(v4f*)(ldsW + row * SWW + c4 * 4) = *(const v4f*)(w + row * D + c4 * 4);
  }
  __syncthreads();

  // Async-issue one channel chunk into buffer p. Per-wave async count is a
  // uniform APT: padded lanes load a clamped valid address into a trash slot.
  auto issue = [&](int c, int p) {
    const int d0 = c * DC;
    float* buf = ldsX + p * XBUF;
    #pragma unroll
    for (int j = 0; j < APT; ++j) {
      int q = tid + 128 * j;        // float4 index, 640 per chunk
      int row = q >> 3;             // 8 float4 per 32-float row
      int c4  = q & 7;
      int g   = i0 + row;
      const float* src = xb + (size_t)(g < S ? g : 0) * D + d0 + c4 * 4;
      float* dst = (g < S) ? (buf + row * SXW + c4 * 4) : ldsTrash;
      ASYNC_COPY_B128(dst, src);
    }
  };

  v8f c0 = {};  // Y rows [i0+16w .. +15]   (this wave's outputs)
  v8f c1 = {};  // Y rows [i0+16w+16 .. +31] (lookahead for the diagonal)
  const int aoff0 = (16 * wave + lm) * SXW + lh * 2;
  const int aoff1 = aoff0 + 16 * SXW;
  const int boff  = lm * SWW + lh * 2;

  issue(0, 0);
  int p = 0;
  for (int c = 0; c < NCH; ++c) {
    if (c + 1 < NCH) { issue(c + 1, p ^ 1); WAIT_ASYNC(APT); }  // retire chunk c only
    else             { WAIT_ASYNC(0); }
    __syncthreads();                       // all waves' chunk-c LDS writes visible
    const float* bufp = ldsX + p * XBUF;
    const int d0 = c * DC;
    #pragma unroll
    for (int s = 0; s < DC / 4; ++s) {
      const int dc = 4 * s;
      v2f bf = *(const v2f*)(ldsW + boff + d0 + dc);   // B[kk,n] = w[n, d0+dc+kk]
      v2f a0 = *(const v2f*)(bufp + aoff0 + dc);       // A[m,kk] = x[i0+16w+m, d0+dc+kk]
      v2f a1 = *(const v2f*)(bufp + aoff1 + dc);
      c0 = __builtin_amdgcn_wmma_f32_16x16x4_f32(false, a0, false, bf,
                                                 (short)0, c0, false, false);
      c1 = __builtin_amdgcn_wmma_f32_16x16x4_f32(false, a1, false, bf,
                                                 (short)0, c1, false, false);
    }
    __syncthreads();                       // readers done: buffer free for chunk c+2
    p ^= 1;
  }

  // Diagonal epilogue: out[i] = bias + sum_k Y[i+k, k]
  float* my = ldsY + wave * 32 * SYW;
  #pragma unroll
  for (int j = 0; j < 8; ++j) {            // C layout: VGPR j -> M=j (lanes<16), M=j+8
    my[(j + 8 * lh) * SYW + lm]      = c0[j];
    my[(16 + j + 8 * lh) * SYW + lm] = c1[j];
  }
  __syncthreads();
  if (lane < 16) {
    float acc = bias[0];
    #pragma unroll
    for (int k = 0; k < 16; ++k) acc += my[(lane + k) * SYW + k];
    out[(size_t)b * S + i0 + 16 * wave + lane] = acc;
  }
}

extern "C" void kernel_launch(void* const* d_in, const int* in_sizes, int n_in,
                              void* d_out, int out_size, void* d_ws, size_t ws_size,
                              hipStream_t stream) {
  const float* x    = (const float*)d_in[0];   // embedded [B,S,D] fp32
  const float* w    = (const float*)d_in[1];   // filt [K*D,1] fp32
  const float* bias = (const float*)d_in[2];   // [1] fp32
  float* out = (float*)d_out;                  // [B,S] fp32

  const int Bn = in_sizes[0] / (S * D);        // 32
  dim3 grid(S / TM, Bn, 1);
  convk16_wmma_f32_async<<<grid, dim3(128, 1, 1), 0, stream>>>(x, w, bias, out);
}